// Action_Decoder_72249939853874
// MI455X (gfx1250) — compile-verified
//
#include <hip/hip_runtime.h>

// ---------------------------------------------------------------------------
// Action decoder: obs projection GEMM (WMMA fp32) + fully fused 2-layer GAT
// over 6-node fully connected graphs. Kernel 2 packs TWO batch elements per
// wave (rows 0-5 and 8-13 of the 16x16 WMMA tile) to raise matrix-unit
// utilization from 37.5% to 75%.
// Target: gfx1250 (CDNA5), wave32, V_WMMA_F32_16X16X4_F32.
// ---------------------------------------------------------------------------

typedef __attribute__((ext_vector_type(2))) float v2f;
typedef __attribute__((ext_vector_type(8))) float v8f;

#define B_    4096
#define N_    177
#define S_    36
#define K_    6
#define H_    128
#define OBS_  500
#define CIN_  384          // 3*H
#define HEADS_ 4
#define FH_    32
#define HROW_  132         // padded LDS row stride (132 % 64 = 4 -> no bank conflicts)
#define W2_   4            // waves per block in kernel 2

__device__ __forceinline__ v8f wmma_f32(v2f a, v2f b, v8f c) {
  // D = A(16x4 f32) * B(4x16 f32) + C(16x16 f32)
  return __builtin_amdgcn_wmma_f32_16x16x4_f32(
      /*neg_a=*/false, a, /*neg_b=*/false, b,
      /*c_mod=*/(short)0, c, /*reuse_a=*/false, /*reuse_b=*/false);
}

// ---------------------------------------------------------------------------
// Kernel 1: obs_repr[4096,128] = org_obs[4096,500] @ W_proj[500,128] + b_proj
// One wave computes a 16x64 tile (4 accumulators). 125 K-steps of WMMA K=4.
// ---------------------------------------------------------------------------
__global__ __launch_bounds__(128) void obs_proj_kernel(
    const float* __restrict__ org_obs, const float* __restrict__ W_proj,
    const float* __restrict__ b_proj, float* __restrict__ obs_repr) {
  const int lane = threadIdx.x & 31;
  const int wave = threadIdx.x >> 5;
  const int gw   = blockIdx.x * 4 + wave;     // 512 waves total
  const int row0 = (gw >> 1) * 16;            // 256 row tiles
  const int col0 = (gw & 1) * 64;             // 2 col halves
  const int m  = lane & 15;                   // A: M index / B: N index
  const int kp = (lane >> 4) * 2;             // A/B: K sub-pair (0 or 2)

  v8f acc[4] = {};
  const float* arow = org_obs + (size_t)(row0 + m) * OBS_;

  for (int k0 = 0; k0 < OBS_; k0 += 4) {      // 500 = 125 * 4, exact
    v2f a;
    a.x = arow[k0 + kp];
    a.y = arow[k0 + kp + 1];
#pragma unroll
    for (int t = 0; t < 4; ++t) {
      v2f bb;
      bb.x = W_proj[(size_t)(k0 + kp)     * H_ + col0 + t * 16 + m];
      bb.y = W_proj[(size_t)(k0 + kp + 1) * H_ + col0 + t * 16 + m];
      acc[t] = wmma_f32(a, bb, acc[t]);
    }
  }

  // C/D layout: VGPR v -> (M = v, N = lane) lanes 0-15, (M = v+8) lanes 16-31.
  const int rhalf = (lane >> 4) * 8;
#pragma unroll
  for (int t = 0; t < 4; ++t) {
#pragma unroll
    for (int v = 0; v < 8; ++v) {
      const int row = row0 + v + rhalf;
      const int col = col0 + t * 16 + m;
      obs_repr[(size_t)row * H_ + col] = acc[t][v] + b_proj[col];
    }
  }
}

// ---------------------------------------------------------------------------
// Kernel 2: fully fused GAT layer 1 (WMMA x@W1 + attention + ELU) and GAT
// layer 2 (scalar head). TWO batches per wave: A-tile rows 0-5 = batch pb=0,
// rows 8-13 = batch pb=1 (matches the M-split of the C/D register layout).
// Gathered A operand (obs | sub | node) is read straight from global; the
// 128-float segments are aligned so the segment branch is wave-uniform.
// ---------------------------------------------------------------------------
__global__ __launch_bounds__(32 * W2_) void gat_fused_kernel(
    const float* __restrict__ node_emb,   // [B, N, H]
    const float* __restrict__ sub_emb,    // [B, S, H]
    const int*   __restrict__ sub_choice, // [B, 1]
    const int*   __restrict__ sub2elem,   // [S, K]
    const float* __restrict__ obs_repr,   // [B, H] (ws)
    const float* __restrict__ W1,         // [384, 128]
    const float* __restrict__ a_src1,     // [4, 32]
    const float* __restrict__ a_dst1,     // [4, 32]
    const float* __restrict__ b1,         // [128]
    const float* __restrict__ W2w,        // [128, 1]
    const float* __restrict__ a_src2,     // [1,1]
    const float* __restrict__ a_dst2,     // [1,1]
    const float* __restrict__ b2,         // [1]
    float* __restrict__ out) {            // [B, 6]
  __shared__ float h_sh [W2_][2][K_ * HROW_];
  __shared__ float h1_sh[W2_][2][K_ * HROW_];
  __shared__ float es_sh[W2_][2][K_ * HEADS_];
  __shared__ float ed_sh[W2_][2][K_ * HEADS_];
  __shared__ float al_sh[W2_][2][K_ * HEADS_ * K_];   // normalized alpha
  __shared__ float s_sh [W2_][2][K_];

  const int lane = threadIdx.x & 31;
  const int wave = threadIdx.x >> 5;
  const int base = (blockIdx.x * W2_ + wave) * 2;     // batches base, base+1

  const int m  = lane & 15;
  const int kp = (lane >> 4) * 2;
  const int bA   = base + (m >> 3);                   // batch owning this A-row
  const int r    = m & 7;                             // local node row
  const int rr   = (r < K_) ? r : (K_ - 1);           // rows 6,7: clamped dupes
  const int subA = sub_choice[bA];
  const int elem = sub2elem[subA * K_ + rr];

  const float* obs_row  = obs_repr + (size_t)bA * H_;
  const float* sub_row  = sub_emb  + ((size_t)bA * S_ + subA) * H_;
  const float* node_row = node_emb + ((size_t)bA * N_ + elem) * H_;

  // ---- GEMM: h[16,128] = x[16,384] @ W1[384,128] (rows 0-5, 8-13 valid) ----
  v8f acc[8] = {};
  for (int k0 = 0; k0 < CIN_; k0 += 4) {              // 96 K-steps
    const int c = k0 + kp;                            // segment is wave-uniform
    v2f a;
    if (c < H_)            { a.x = obs_row[c];            a.y = obs_row[c + 1]; }
    else if (c < 2 * H_)   { a.x = sub_row[c - H_];       a.y = sub_row[c - H_ + 1]; }
    else                   { a.x = node_row[c - 2 * H_];  a.y = node_row[c - 2 * H_ + 1]; }
#pragma unroll
    for (int t = 0; t < 8; ++t) {
      v2f bb;
      bb.x = W1[(size_t)c       * (HEADS_ * FH_) + t * 16 + m];
      bb.y = W1[(size_t)(c + 1) * (HEADS_ * FH_) + t * 16 + m];
      acc[t] = wmma_f32(a, bb, acc[t]);
    }
  }

  // Spill valid h rows to LDS. C layout: VGPR v = (M=v, lanes 0-15) and
  // (M=v+8, lanes 16-31) -> lanes 0-15 carry batch0 rows, 16-31 batch1 rows.
  const int pbC = lane >> 4;
#pragma unroll
  for (int t = 0; t < 8; ++t) {
#pragma unroll
    for (int v = 0; v < 8; ++v) {
      if (v < K_) h_sh[wave][pbC][v * HROW_ + t * 16 + m] = acc[t][v];
    }
  }
  __syncthreads();

  // ---- attention logits e_src/e_dst [2 batches x 6 nodes x 4 heads] ----
#pragma unroll
  for (int pb = 0; pb < 2; ++pb) {
    if (lane < K_ * HEADS_) {
      const int j  = lane >> 2;
      const int hd = lane & 3;
      float es = 0.f, ed = 0.f;
      for (int f = 0; f < FH_; ++f) {
        const float hv = h_sh[wave][pb][j * HROW_ + hd * FH_ + f];
        es += hv * a_src1[hd * FH_ + f];
        ed += hv * a_dst1[hd * FH_ + f];
      }
      es_sh[wave][pb][lane] = es;
      ed_sh[wave][pb][lane] = ed;
    }
  }
  __syncthreads();

  // ---- cooperative softmax: one lane per (i,head) computes normalized alpha
#pragma unroll
  for (int pb = 0; pb < 2; ++pb) {
    if (lane < K_ * HEADS_) {
      const int i  = lane >> 2;
      const int hd = lane & 3;
      const float edv = ed_sh[wave][pb][i * HEADS_ + hd];
      float w[K_];
      float mx = -3.4e38f;
#pragma unroll
      for (int j = 0; j < K_; ++j) {
        float e = edv + es_sh[wave][pb][j * HEADS_ + hd];
        e = (e > 0.f) ? e : 0.2f * e;                 // leaky_relu(0.2)
        w[j] = e;
        mx = fmaxf(mx, e);
      }
      float sum = 0.f;
#pragma unroll
      for (int j = 0; j < K_; ++j) { w[j] = expf(w[j] - mx); sum += w[j]; }
      const float inv = 1.f / sum;
#pragma unroll
      for (int j = 0; j < K_; ++j)
        al_sh[wave][pb][lane * K_ + j] = w[j] * inv;
    }
  }
  __syncthreads();

  // ---- combine: h1[i,col] = ELU(sum_j alpha*h[j,col] + b1); lane = f ----
#pragma unroll
  for (int pb = 0; pb < 2; ++pb) {
#pragma unroll
    for (int i = 0; i < K_; ++i) {
#pragma unroll
      for (int hd = 0; hd < HEADS_; ++hd) {
        const int col = hd * FH_ + lane;
        float val = 0.f;
#pragma unroll
        for (int j = 0; j < K_; ++j)
          val += al_sh[wave][pb][(i * HEADS_ + hd) * K_ + j] *
                 h_sh[wave][pb][j * HROW_ + col];
        val += b1[col];
        val = (val > 0.f) ? val : (expf(val) - 1.f);  // ELU(alpha=1)
        h1_sh[wave][pb][i * HROW_ + col] = val;
      }
    }
  }
  __syncthreads();

  // ---- layer 2: s[i] = h1[i] . W2 (12 rows across the wave) ----
  if (lane < 2 * K_) {
    const int pb = lane / K_;
    const int i  = lane % K_;
    float s = 0.f;
    for (int c = 0; c < H_; ++c) s += h1_sh[wave][pb][i * HROW_ + c] * W2w[c];
    s_sh[wave][pb][i] = s;
  }
  __syncthreads();

  // ---- layer 2 attention (heads=1, fh=1) + output ----
  if (lane < 2 * K_) {
    const int pb = lane / K_;
    const int i  = lane % K_;
    const float as2 = a_src2[0], ad2 = a_dst2[0];
    const float edv = s_sh[wave][pb][i] * ad2;
    float w[K_], mx = -3.4e38f;
#pragma unroll
    for (int j = 0; j < K_; ++j) {
      float e = edv + s_sh[wave][pb][j] * as2;
      e = (e > 0.f) ? e : 0.2f * e;
      w[j] = e;
      mx = fmaxf(mx, e);
    }
    float sum = 0.f;
#pragma unroll
    for (int j = 0; j < K_; ++j) { w[j] = expf(w[j] - mx); sum += w[j]; }
    float val = 0.f;
#pragma unroll
    for (int j = 0; j < K_; ++j) val += w[j] * s_sh[wave][pb][j];
    out[(size_t)(base + pb) * K_ + i] = val / sum + b2[0];
  }
}

// ---------------------------------------------------------------------------
extern "C" void kernel_launch(void* const* d_in, const int* in_sizes, int n_in,
                              void* d_out, int out_size, void* d_ws, size_t ws_size,
                              hipStream_t stream) {
  const float* org_obs    = (const float*)d_in[0];
  const float* node_emb   = (const float*)d_in[1];
  const float* sub_emb    = (const float*)d_in[2];
  const int*   sub_choice = (const int*)  d_in[3];
  const int*   sub2elem   = (const int*)  d_in[4];
  const float* W_proj     = (const float*)d_in[5];
  const float* b_proj     = (const float*)d_in[6];
  const float* W1         = (const float*)d_in[7];
  const float* a_src1     = (const float*)d_in[8];
  const float* a_dst1     = (const float*)d_in[9];
  const float* b1         = (const float*)d_in[10];
  const float* W2w        = (const float*)d_in[11];
  const float* a_src2     = (const float*)d_in[12];
  const float* a_dst2     = (const float*)d_in[13];
  const float* b2         = (const float*)d_in[14];
  float* out      = (float*)d_out;
  float* obs_repr = (float*)d_ws;   // 4096*128 floats = 2 MB scratch

  // Kernel 1: 512 waves (256 row tiles x 2 col halves), 4 waves/block.
  obs_proj_kernel<<<128, 128, 0, stream>>>(org_obs, W_proj, b_proj, obs_repr);

  // Kernel 2: two batches per wave, 4 waves/block -> 4096/(2*4) = 512 blocks.
  gat_fused_kernel<<<512, 32 * W2_, 0, stream>>>(node_emb, sub_emb, sub_choice,
      sub2elem, obs_repr, W1, a_src1, a_dst1, b1, W2w, a_src2, a_dst2, b2, out);
}